// TraverseSubGNN_10746008174933
// MI455X (gfx1250) — compile-verified
//
#include <hip/hip_runtime.h>
#include <hip/hip_bf16.h>

typedef __attribute__((ext_vector_type(2))) float v2f;
typedef __attribute__((ext_vector_type(8))) float v8f;

static constexpr int LAYERS = 3;
static constexpr int SUB    = 1000;
static constexpr int IN_F   = 8;
static constexpr int OUT_F  = 48;
static constexpr int HEADS  = 3;

__device__ __forceinline__ float lrelu(float v, float s) { return v > 0.f ? v : v * s; }

// float atomic max via sign-split trick (valid with -inf init, mixed signs)
__device__ __forceinline__ void atomicMaxF(float* a, float v) {
    if (v >= 0.f) atomicMax((int*)a, __float_as_int(v));
    else          atomicMin((unsigned int*)a, __float_as_uint(v));
}

// ---------------- init scratch ----------------
__global__ void k_init(float* __restrict__ aggr, float* __restrict__ amax,
                       float* __restrict__ denom, float* __restrict__ gsum,
                       float* __restrict__ gsq, int n) {
    int i = blockIdx.x * blockDim.x + threadIdx.x;
    if (i < n * OUT_F) aggr[i] = 0.f;
    if (i < n * 4) { amax[i] = __int_as_float(0xff800000); denom[i] = 0.f; }  // -inf / 0
    if (i < OUT_F) { gsum[i] = 0.f; gsq[i] = 0.f; }
}

// ---------------- fold att into the GEMM: P[col][j], col 0..2 = dst heads, 3..5 = src ----
__global__ void k_prep(const float* __restrict__ W, const float* __restrict__ att,
                       float* __restrict__ P) {
    int t = threadIdx.x;                 // 48 threads: col = t/8 in [0,6), j = t%8
    if (t >= 6 * IN_F) return;
    int col = t / IN_F;
    int j   = t % IN_F;
    int h   = col % HEADS;
    int off = (col >= HEADS) ? 16 : 0;   // dst uses att[h,:16], src uses att[h,16:]
    float acc = 0.f;
#pragma unroll
    for (int k = 0; k < 16; ++k)
        acc += W[(size_t)(h * 16 + k) * IN_F + j] * att[h * 32 + off + k];
    P[col * IN_F + j] = acc;
}

// ---------------- x = subx @ W.T (+ fused s_dst/s_src columns), all via WMMA f32 ------
__global__ __launch_bounds__(256) void k_feat(const float* __restrict__ subx,
                                              const float* __restrict__ W,
                                              const float* __restrict__ P,
                                              float* __restrict__ x,
                                              float* __restrict__ sdst,
                                              float* __restrict__ ssrc, int n) {
    const int lane = threadIdx.x & 31;
    const int wave = threadIdx.x >> 5;
    const int tile = blockIdx.x * 8 + wave;
    const int row0 = tile * 16;
    if (row0 >= n) return;                       // wave-uniform: EXEC all-ones for WMMA
    const int m    = lane & 15;
    const int half = lane >> 4;
    const bool full = (row0 + 16 <= n);

    // A fragment 16x4 f32: VGPR0 = K{0|2}, VGPR1 = K{1|3} per half (k-steps k0=0,4)
    int rowA = row0 + m; if (rowA >= n) rowA = n - 1;
    const float* ap = subx + (size_t)rowA * IN_F;
    v2f a0, a1;
    a0.x = ap[2 * half + 0]; a0.y = ap[2 * half + 1];
    a1.x = ap[4 + 2 * half + 0]; a1.y = ap[4 + 2 * half + 1];

    // --- three 16-column tiles of x (head h == column tile h) ---
#pragma unroll
    for (int h = 0; h < HEADS; ++h) {
        const int col = h * 16 + m;              // B[k][col] = W[col][k]
        const float* wp = W + (size_t)col * IN_F;
        v2f b0, b1;
        b0.x = wp[2 * half]; b0.y = wp[2 * half + 1];
        b1.x = wp[4 + 2 * half]; b1.y = wp[4 + 2 * half + 1];

        v8f acc = {};
        acc = __builtin_amdgcn_wmma_f32_16x16x4_f32(false, a0, false, b0, (short)0, acc, false, false);
        acc = __builtin_amdgcn_wmma_f32_16x16x4_f32(false, a1, false, b1, (short)0, acc, false, false);

        float* xp = x + (size_t)row0 * OUT_F + col;
        if (full) {
#pragma unroll
            for (int r = 0; r < 8; ++r)
                xp[(size_t)(r + 8 * half) * OUT_F] = acc[r];
        } else {
#pragma unroll
            for (int r = 0; r < 8; ++r)
                if (row0 + r + 8 * half < n) xp[(size_t)(r + 8 * half) * OUT_F] = acc[r];
        }
    }

    // --- 4th tile: attention projections. B cols 0..2 = P_dst, 3..5 = P_src, rest 0 ---
    {
        v2f b0 = {0.f, 0.f}, b1 = {0.f, 0.f};
        if (m < 6) {
            const float* pp = P + m * IN_F;
            b0.x = pp[2 * half]; b0.y = pp[2 * half + 1];
            b1.x = pp[4 + 2 * half]; b1.y = pp[4 + 2 * half + 1];
        }
        v8f acc = {};
        acc = __builtin_amdgcn_wmma_f32_16x16x4_f32(false, a0, false, b0, (short)0, acc, false, false);
        acc = __builtin_amdgcn_wmma_f32_16x16x4_f32(false, a1, false, b1, (short)0, acc, false, false);
#pragma unroll
        for (int r = 0; r < 8; ++r) {
            const int row = row0 + r + 8 * half;
            if (row < n) {
                if (m < HEADS)                sdst[(size_t)row * 4 + m]         = acc[r];
                else if (m < 2 * HEADS)       ssrc[(size_t)row * 4 + (m - 3)]   = acc[r];
            }
        }
    }
}

// ---------------- edge pass 1: segment max over src ----------------
__global__ void k_edge_max(const int* __restrict__ ei, const float4* __restrict__ sdst,
                           const float4* __restrict__ ssrc, float* __restrict__ amax, int E) {
    int e = blockIdx.x * blockDim.x + threadIdx.x;
    if (e >= E) return;
    int src = ei[e], dst = ei[E + e];
    float4 sd = sdst[dst], ss = ssrc[src];
    atomicMaxF(&amax[src * 4 + 0], lrelu(sd.x + ss.x, 0.2f));
    atomicMaxF(&amax[src * 4 + 1], lrelu(sd.y + ss.y, 0.2f));
    atomicMaxF(&amax[src * 4 + 2], lrelu(sd.z + ss.z, 0.2f));
}

// ---------------- edge pass 2: segment exp-sum over src ----------------
__global__ void k_edge_sum(const int* __restrict__ ei, const float4* __restrict__ sdst,
                           const float4* __restrict__ ssrc, const float4* __restrict__ amax,
                           float* __restrict__ denom, int E) {
    int e = blockIdx.x * blockDim.x + threadIdx.x;
    if (e >= E) return;
    int src = ei[e], dst = ei[E + e];
    float4 sd = sdst[dst], ss = ssrc[src], am = amax[src];
    atomicAdd(&denom[src * 4 + 0], expf(lrelu(sd.x + ss.x, 0.2f) - am.x));
    atomicAdd(&denom[src * 4 + 1], expf(lrelu(sd.y + ss.y, 0.2f) - am.y));
    atomicAdd(&denom[src * 4 + 2], expf(lrelu(sd.z + ss.z, 0.2f) - am.z));
}

// ---------------- edge pass 3: weighted scatter-add of masked x_j ----------------
__global__ void k_edge_aggr(const int* __restrict__ ei, const int* __restrict__ mask,
                            const float4* __restrict__ sdst, const float4* __restrict__ ssrc,
                            const float4* __restrict__ amax, const float4* __restrict__ denom,
                            const float4* __restrict__ x4, float* __restrict__ aggr, int E) {
    int e = blockIdx.x * blockDim.x + threadIdx.x;
    if (e >= E) return;
    if (mask[e] == 0) return;                    // masked edge contributes nothing
    int src = ei[e], dst = ei[E + e];
    float4 sd = sdst[dst], ss = ssrc[src], am = amax[src], dn = denom[src];
    float w[3];
    w[0] = expf(lrelu(sd.x + ss.x, 0.2f) - am.x) / (dn.x + 1e-16f);
    w[1] = expf(lrelu(sd.y + ss.y, 0.2f) - am.y) / (dn.y + 1e-16f);
    w[2] = expf(lrelu(sd.z + ss.z, 0.2f) - am.z) / (dn.z + 1e-16f);
    const float4* xs = x4 + (size_t)src * 12;    // 48 floats = 12 float4, 4 per head
    float* out = aggr + (size_t)dst * OUT_F;
#pragma unroll
    for (int i = 0; i < 12; ++i) {
        float4 v = xs[i];
        float wh = w[i >> 2];
        atomicAdd(out + 4 * i + 0, v.x * wh);
        atomicAdd(out + 4 * i + 1, v.y * wh);
        atomicAdd(out + 4 * i + 2, v.z * wh);
        atomicAdd(out + 4 * i + 3, v.w * wh);
    }
}

// ---------------- BN stats: per-block LDS accumulation, then global atomics ----------
__global__ __launch_bounds__(256) void k_stats(const float* __restrict__ aggr,
                                               const float* __restrict__ bias,
                                               float* __restrict__ gsum,
                                               float* __restrict__ gsq, int n) {
    __shared__ float ls[OUT_F], lq[OUT_F];
    if (threadIdx.x < OUT_F) { ls[threadIdx.x] = 0.f; lq[threadIdx.x] = 0.f; }
    __syncthreads();
    const size_t total = (size_t)n * OUT_F;
    const size_t base  = (size_t)blockIdx.x * 256 * OUT_F;
#pragma unroll 4
    for (int it = 0; it < OUT_F; ++it) {
        size_t idx = base + (size_t)it * 256 + threadIdx.x;
        if (idx < total) {
            int c = (int)(idx % OUT_F);
            float y = lrelu(aggr[idx] + bias[c], 0.1f);
            atomicAdd(&ls[c], y);
            atomicAdd(&lq[c], y * y);
        }
    }
    __syncthreads();
    if (threadIdx.x < OUT_F) {
        atomicAdd(&gsum[threadIdx.x], ls[threadIdx.x]);
        atomicAdd(&gsq[threadIdx.x],  lq[threadIdx.x]);
    }
}

// ---------------- fused bias -> lrelu -> BN -> hop mean ----------------
__global__ void k_final(const float* __restrict__ aggr, const float* __restrict__ bias,
                        const float* __restrict__ gamma, const float* __restrict__ beta,
                        const float* __restrict__ gsum, const float* __restrict__ gsq,
                        float* __restrict__ out, int n, int outSize) {
    int i = blockIdx.x * blockDim.x + threadIdx.x;
    if (i >= outSize) return;
    int c = i % OUT_F;
    int t = i / OUT_F;
    int s = t % SUB;
    int b = t / SUB;
    float invN = 1.f / (float)n;
    float mean = gsum[c] * invN;
    float var  = gsq[c] * invN - mean * mean;
    float inv  = rsqrtf(var + 1e-5f);
    float g = gamma[c], bb = beta[c], bi = bias[c];
    float acc = 0.f;
#pragma unroll
    for (int l = 0; l < LAYERS; ++l) {
        int node = (b * LAYERS + l) * SUB + s;
        float y = lrelu(aggr[(size_t)node * OUT_F + c] + bi, 0.1f);
        acc += g * (y - mean) * inv + bb;
    }
    out[i] = acc * (1.f / (float)LAYERS);
}

extern "C" void kernel_launch(void* const* d_in, const int* in_sizes, int n_in,
                              void* d_out, int out_size, void* d_ws, size_t ws_size,
                              hipStream_t stream) {
    const float* subx  = (const float*)d_in[0];
    const int*   ei    = (const int*)d_in[1];
    const int*   mask  = (const int*)d_in[2];
    const float* W     = (const float*)d_in[3];
    const float* att   = (const float*)d_in[4];
    const float* bias  = (const float*)d_in[5];
    const float* gamma = (const float*)d_in[6];
    const float* beta  = (const float*)d_in[7];

    const int N = in_sizes[0] / IN_F;   // 96000
    const int E = in_sizes[1] / 2;      // 1,536,000

    // workspace layout (floats), all sections 64B-aligned
    float* ws    = (float*)d_ws;
    float* x     = ws;                          // N*48
    float* sdst  = x    + (size_t)N * OUT_F;    // N*4
    float* ssrc  = sdst + (size_t)N * 4;        // N*4
    float* amax  = ssrc + (size_t)N * 4;        // N*4
    float* denom = amax + (size_t)N * 4;        // N*4
    float* aggr  = denom + (size_t)N * 4;       // N*48
    float* gsum  = aggr + (size_t)N * OUT_F;    // 48
    float* gsq   = gsum + OUT_F;                // 48
    float* P     = gsq  + OUT_F;                // 48 (8x6 folded att projection)

    const int T = 256;

    k_init<<<(N * OUT_F + T - 1) / T, T, 0, stream>>>(aggr, amax, denom, gsum, gsq, N);
    k_prep<<<1, 64, 0, stream>>>(W, att, P);

    const int nTiles = (N + 15) / 16;
    k_feat<<<(nTiles + 7) / 8, T, 0, stream>>>(subx, W, P, x, sdst, ssrc, N);

    const int eBlocks = (E + T - 1) / T;
    k_edge_max<<<eBlocks, T, 0, stream>>>(ei, (const float4*)sdst, (const float4*)ssrc, amax, E);
    k_edge_sum<<<eBlocks, T, 0, stream>>>(ei, (const float4*)sdst, (const float4*)ssrc,
                                          (const float4*)amax, denom, E);
    k_edge_aggr<<<eBlocks, T, 0, stream>>>(ei, mask, (const float4*)sdst, (const float4*)ssrc,
                                           (const float4*)amax, (const float4*)denom,
                                           (const float4*)x, aggr, E);

    k_stats<<<(N + T - 1) / T, T, 0, stream>>>(aggr, bias, gsum, gsq, N);

    k_final<<<(out_size + T - 1) / T, T, 0, stream>>>(aggr, bias, gamma, beta, gsum, gsq,
                                                      (float*)d_out, N, out_size);
}